// MixtureModelDecoder_8495445311672
// MI455X (gfx1250) — compile-verified
//
#include <hip/hip_runtime.h>
#include <hip/hip_bf16.h>
#include <math.h>

// ---------------------------------------------------------------------------
// MixtureModelDecoder fused kernel for gfx1250 (CDNA5, wave32, WMMA).
//
// One workgroup (256 threads = 8 wave32) per batch element b (grid = 1024):
//   1) cluster-head resblock (eta 64 -> 128 relu -> 32) + log-softmax  (VALU)
//   2) per-node argmax(log_pi + gumbel) -> one-hot gather of cluster
//      means / log-sigmas -> z = z_noise * exp(clip(ls)) + mu  (c==y_hard
//      exactly since y_soft - stop_grad(y_soft) == 0)
//   3) G_t = Z * Wsym_t * Z^T  per t (5x 40x40) via V_WMMA_F32_16X16X4_F32,
//      all tiles resident in LDS (Z padded to 48 rows)
//   4) atom-head resblock (z 32 -> 128 relu -> 10)                    (VALU)
//   5) edge head: bil[e,t] = G_t[i,j] gather, resblock 5->64->5,
//      softmax over 5                                                 (VALU)
// ---------------------------------------------------------------------------

typedef float v2f __attribute__((ext_vector_type(2)));
typedef float v8f __attribute__((ext_vector_type(8)));

#define NB_  5
#define NN_  40
#define KK_  32
#define ZD_  32
#define ETA_ 64
#define CH_  128
#define AH_  128
#define NA_  10
#define BH_  64
#define NPAIR_ 780   // C(40,2)

__global__ __launch_bounds__(256)
void mmdecoder_fused(const float* __restrict__ eta,
                     const float* __restrict__ gumbel,
                     const float* __restrict__ z_noise,
                     const float* __restrict__ cW1, const float* __restrict__ cb1,
                     const float* __restrict__ cW2, const float* __restrict__ cb2,
                     const float* __restrict__ cWs, const float* __restrict__ cbs,
                     const float* __restrict__ cmeans,
                     const float* __restrict__ clsig,
                     const float* __restrict__ aW1, const float* __restrict__ ab1,
                     const float* __restrict__ aW2, const float* __restrict__ ab2,
                     const float* __restrict__ aWs, const float* __restrict__ abs_,
                     const float* __restrict__ bond,
                     const float* __restrict__ bW1, const float* __restrict__ bb1,
                     const float* __restrict__ bW2, const float* __restrict__ bb2,
                     const float* __restrict__ bWs, const float* __restrict__ bbs,
                     const int*   __restrict__ eidx,
                     float* __restrict__ out_atom,   // (B*40, 10)
                     float* __restrict__ out_edge)   // (B*780, 5)
{
    const int b    = blockIdx.x;
    const int tid  = threadIdx.x;
    const int lane = tid & 31;
    const int wave = tid >> 5;           // 8 waves of 32

    __shared__ float etas[ETA_];
    __shared__ float h1s[CH_];
    __shared__ float logp[KK_];
    __shared__ float s_lse;
    __shared__ int   kst[NN_];
    __shared__ float Zs[48][ZD_];        // z, zero-padded rows 40..47
    __shared__ float Ys[48][ZD_];        // Z @ Wsym_t (reused per t)
    __shared__ float Gs[NB_][NN_][NN_];  // bilinear forms, 32 KB
    __shared__ float hb[8][AH_];         // atom-head hidden, per wave
    __shared__ float s_bW1[NB_*BH_], s_bW2[BH_*NB_], s_bWs[NB_*NB_];
    __shared__ float s_bb1[BH_], s_bb2[NB_], s_bbs[NB_];

    // ---- stage edge-head weights into LDS ---------------------------------
    for (int i = tid; i < NB_*BH_; i += 256) s_bW1[i] = bW1[i];
    for (int i = tid; i < BH_*NB_; i += 256) s_bW2[i] = bW2[i];
    if (tid < NB_*NB_) s_bWs[tid] = bWs[tid];
    if (tid < BH_)     s_bb1[tid] = bb1[tid];
    if (tid < NB_)   { s_bb2[tid] = bb2[tid]; s_bbs[tid] = bbs[tid]; }
    if (tid < ETA_)    etas[tid]  = eta[b*ETA_ + tid];
    __syncthreads();

    // ---- 1) cluster head: resblock + log-softmax --------------------------
    if (tid < CH_) {
        float acc = cb1[tid];
        for (int i = 0; i < ETA_; ++i) acc = fmaf(etas[i], cW1[i*CH_ + tid], acc);
        h1s[tid] = fmaxf(acc, 0.0f);
    }
    __syncthreads();
    if (tid < KK_) {
        float acc = cb2[tid] + cbs[tid];
        for (int j = 0; j < CH_;  ++j) acc = fmaf(h1s[j],  cW2[j*KK_ + tid], acc);
        for (int i = 0; i < ETA_; ++i) acc = fmaf(etas[i], cWs[i*KK_ + tid], acc);
        logp[tid] = acc;
    }
    __syncthreads();
    if (tid == 0) {
        float m = logp[0];
        for (int k = 1; k < KK_; ++k) m = fmaxf(m, logp[k]);
        float s = 0.0f;
        for (int k = 0; k < KK_; ++k) s += __expf(logp[k] - m);
        s_lse = m + __logf(s);
    }
    __syncthreads();
    if (tid < KK_) logp[tid] = logp[tid] - s_lse;   // log_pi[b, :]
    __syncthreads();

    // ---- 2) per-node hard assignment + z ----------------------------------
    if (tid < NN_) {
        const float* g = gumbel + (size_t)(b*NN_ + tid) * KK_;
        float best = logp[0] + g[0]; int bi = 0;
        for (int k = 1; k < KK_; ++k) {
            float v = logp[k] + g[k];
            if (v > best) { best = v; bi = k; }
        }
        kst[tid] = bi;
    }
    __syncthreads();
    for (int idx = tid; idx < 48*ZD_; idx += 256) {
        const int n = idx >> 5, k = idx & 31;
        float zv = 0.0f;
        if (n < NN_) {
            const int ks = kst[n];
            const float mu  = cmeans[ks*ZD_ + k];
            float ls = clsig[ks*ZD_ + k];
            ls = fminf(fmaxf(ls, -20.0f), 30.0f);
            zv = fmaf(z_noise[(size_t)(b*NN_ + n)*ZD_ + k], __expf(ls), mu);
        }
        Zs[n][k] = zv;
    }
    __syncthreads();

    // ---- 3) G_t = Z Wsym_t Z^T via fp32 WMMA ------------------------------
    // A 16x4 frag: lane l -> row m=l&15; VGPR0/1 hold K = 2*(l>>4) + {0,1}
    // B  4x16 frag: lane l -> col n=l&15; VGPR0/1 hold K = 2*(l>>4) + {0,1}
    // C/D 16x16  : VGPR r  -> row m = r + 8*(l>>4), col n = l&15
    const int lh = lane & 15;
    const int kh = (lane >> 4) * 2;

    for (int t = 0; t < NB_; ++t) {
        const float* bt = bond + t*ZD_*ZD_;
        // -- Y = Z @ Wsym_t : tiles (mt in 0..2) x (nt in 0..1), K = 32
        for (int job = wave; job < 6; job += 8) {
            const int m0 = (job >> 1) * 16, n0 = (job & 1) * 16;
            v8f c = {0.f,0.f,0.f,0.f,0.f,0.f,0.f,0.f};
            for (int k0 = 0; k0 < ZD_; k0 += 4) {
                v2f a, bf;
                a[0] = Zs[m0 + lh][k0 + kh + 0];
                a[1] = Zs[m0 + lh][k0 + kh + 1];
                const int kk = k0 + kh, nn = n0 + lh;
                bf[0] = 0.5f * (bt[(kk+0)*ZD_ + nn] + bt[nn*ZD_ + (kk+0)]);
                bf[1] = 0.5f * (bt[(kk+1)*ZD_ + nn] + bt[nn*ZD_ + (kk+1)]);
                c = __builtin_amdgcn_wmma_f32_16x16x4_f32(
                        false, a, false, bf, (short)0, c, false, false);
            }
            for (int r = 0; r < 8; ++r)
                Ys[m0 + r + 8*(lane>>4)][n0 + lh] = c[r];
        }
        __syncthreads();
        // -- G_t = Y @ Z^T : tiles (mt in 0..2) x (nt in 0..2), K = 32
        for (int job = wave; job < 9; job += 8) {
            const int m0 = (job / 3) * 16, n0 = (job % 3) * 16;
            v8f c = {0.f,0.f,0.f,0.f,0.f,0.f,0.f,0.f};
            for (int k0 = 0; k0 < ZD_; k0 += 4) {
                v2f a, bf;
                a[0] = Ys[m0 + lh][k0 + kh + 0];
                a[1] = Ys[m0 + lh][k0 + kh + 1];
                bf[0] = Zs[n0 + lh][k0 + kh + 0];   // B[k][n] = Z[n][k]
                bf[1] = Zs[n0 + lh][k0 + kh + 1];
                c = __builtin_amdgcn_wmma_f32_16x16x4_f32(
                        false, a, false, bf, (short)0, c, false, false);
            }
            for (int r = 0; r < 8; ++r) {
                const int m = m0 + r + 8*(lane>>4);
                const int n = n0 + lh;
                if (m < NN_ && n < NN_) Gs[t][m][n] = c[r];
            }
        }
        __syncthreads();
    }

    // ---- 4) atom head: resblock(z; 32->128 relu ->10) ---------------------
    for (int n = wave; n < NN_; n += 8) {          // uniform trip count (5)
        for (int rep = 0; rep < 4; ++rep) {
            const int jj = lane + rep*32;
            float acc = ab1[jj];
            for (int d = 0; d < ZD_; ++d)
                acc = fmaf(Zs[n][d], aW1[d*AH_ + jj], acc);
            hb[wave][jj] = fmaxf(acc, 0.0f);
        }
        __syncthreads();
        if (lane < NA_) {
            float acc = ab2[lane] + abs_[lane];
            for (int jj = 0; jj < AH_; ++jj)
                acc = fmaf(hb[wave][jj], aW2[jj*NA_ + lane], acc);
            for (int d = 0; d < ZD_; ++d)
                acc = fmaf(Zs[n][d], aWs[d*NA_ + lane], acc);
            out_atom[((size_t)b*NN_ + n)*NA_ + lane] = acc;
        }
        __syncthreads();
    }

    // ---- 5) edge head: gather G, resblock 5->64->5, softmax ---------------
    for (int p = tid; p < NPAIR_; p += 256) {
        const size_t e = (size_t)b*NPAIR_ + p;
        const int i = eidx[2*e + 0] - b*NN_;
        const int j = eidx[2*e + 1] - b*NN_;
        float bil[NB_];
        for (int t = 0; t < NB_; ++t) bil[t] = Gs[t][i][j];

        float acc[NB_];
        for (int oc = 0; oc < NB_; ++oc) {
            float a = s_bb2[oc] + s_bbs[oc];
            for (int t = 0; t < NB_; ++t) a = fmaf(bil[t], s_bWs[t*NB_ + oc], a);
            acc[oc] = a;
        }
        for (int jj = 0; jj < BH_; ++jj) {
            float hv = s_bb1[jj];
            for (int t = 0; t < NB_; ++t) hv = fmaf(bil[t], s_bW1[t*BH_ + jj], hv);
            hv = fmaxf(hv, 0.0f);
            for (int oc = 0; oc < NB_; ++oc) acc[oc] = fmaf(hv, s_bW2[jj*NB_ + oc], acc[oc]);
        }
        float m = acc[0];
        for (int oc = 1; oc < NB_; ++oc) m = fmaxf(m, acc[oc]);
        float s = 0.0f, ex[NB_];
        for (int oc = 0; oc < NB_; ++oc) { ex[oc] = __expf(acc[oc] - m); s += ex[oc]; }
        const float inv = 1.0f / s;
        for (int oc = 0; oc < NB_; ++oc) out_edge[e*NB_ + oc] = ex[oc] * inv;
    }
}

extern "C" void kernel_launch(void* const* d_in, const int* in_sizes, int n_in,
                              void* d_out, int out_size, void* d_ws, size_t ws_size,
                              hipStream_t stream) {
    (void)in_sizes; (void)n_in; (void)d_ws; (void)ws_size; (void)out_size;
    const float* eta     = (const float*)d_in[0];
    const float* gumbel  = (const float*)d_in[1];
    const float* z_noise = (const float*)d_in[2];
    const float* cW1     = (const float*)d_in[3];
    const float* cb1     = (const float*)d_in[4];
    const float* cW2     = (const float*)d_in[5];
    const float* cb2     = (const float*)d_in[6];
    const float* cWs     = (const float*)d_in[7];
    const float* cbs     = (const float*)d_in[8];
    const float* cmeans  = (const float*)d_in[9];
    const float* clsig   = (const float*)d_in[10];
    const float* aW1     = (const float*)d_in[11];
    const float* ab1     = (const float*)d_in[12];
    const float* aW2     = (const float*)d_in[13];
    const float* ab2     = (const float*)d_in[14];
    const float* aWs     = (const float*)d_in[15];
    const float* abs_    = (const float*)d_in[16];
    const float* bond    = (const float*)d_in[17];
    const float* bW1     = (const float*)d_in[18];
    const float* bb1     = (const float*)d_in[19];
    const float* bW2     = (const float*)d_in[20];
    const float* bb2     = (const float*)d_in[21];
    const float* bWs     = (const float*)d_in[22];
    const float* bbs     = (const float*)d_in[23];
    const int*   eidx    = (const int*)  d_in[24];

    float* out_atom = (float*)d_out;                       // 1024*40*10
    float* out_edge = (float*)d_out + 1024*40*10;          // 1024*780*5

    hipLaunchKernelGGL(mmdecoder_fused, dim3(1024), dim3(256), 0, stream,
                       eta, gumbel, z_noise,
                       cW1, cb1, cW2, cb2, cWs, cbs,
                       cmeans, clsig,
                       aW1, ab1, aW2, ab2, aWs, abs_,
                       bond, bW1, bb1, bW2, bb2, bWs, bbs,
                       eidx, out_atom, out_edge);
}